// Smodel_15247133901692
// MI455X (gfx1250) — compile-verified
//
#include <hip/hip_runtime.h>

typedef float v2f __attribute__((ext_vector_type(2)));
typedef float v4f __attribute__((ext_vector_type(4)));
typedef float v8f __attribute__((ext_vector_type(8)));

#define HDIM 64
#define EPSV 1e-5f
// swizzled W in LDS: row-pair k2 holds 64 cols * 2 dwords, padded to 160 dwords
// so hi=0 (k2 even) and hi=1 (k2 odd) halves hit disjoint 32-bank sets.
#define WSTRIDE 160

// ---------------------------------------------------------------------------
// WMMA: D(16x16,f32) = A(16x4,f32) * B(4x16,f32) + C
// ---------------------------------------------------------------------------
__device__ __forceinline__ v8f wmma4(v2f a, v2f b, v8f c) {
  return __builtin_amdgcn_wmma_f32_16x16x4_f32(false, a, false, b, (short)0, c,
                                               false, false);
}

// ---------------------------------------------------------------------------
// Geometry features: dij, djk, theta per triplet
// ---------------------------------------------------------------------------
__global__ void k_geo_feats(const float* __restrict__ coords,
                            const int* __restrict__ ti, const int* __restrict__ tj,
                            const int* __restrict__ tk,
                            float* __restrict__ geo3, int T) {
  int r = blockIdx.x * blockDim.x + threadIdx.x;
  if (r >= T) return;
  int i = ti[r], j = tj[r], k = tk[r];
  float cix = coords[3 * i], ciy = coords[3 * i + 1], ciz = coords[3 * i + 2];
  float cjx = coords[3 * j], cjy = coords[3 * j + 1], cjz = coords[3 * j + 2];
  float ckx = coords[3 * k], cky = coords[3 * k + 1], ckz = coords[3 * k + 2];
  float v1x = cjx - cix, v1y = cjy - ciy, v1z = cjz - ciz;
  float v2x = ckx - cjx, v2y = cky - cjy, v2z = ckz - cjz;
  float dij = sqrtf(v1x * v1x + v1y * v1y + v1z * v1z);
  float wx = cjx - ckx, wy = cjy - cky, wz = cjz - cky - (cjz - ckz) + (cjz - ckz);
  wy = cjy - cky; wz = cjz - ckz;
  float djk = sqrtf(wx * wx + wy * wy + wz * wz);
  float cx = v1y * v2z - v1z * v2y;
  float cy = v1z * v2x - v1x * v2z;
  float cz = v1x * v2y - v1y * v2x;
  float cn = sqrtf(cx * cx + cy * cy + cz * cz);
  float dt = v1x * v2x + v1y * v2y + v1z * v2z;
  geo3[3 * r + 0] = dij;
  geo3[3 * r + 1] = djk;
  geo3[3 * r + 2] = atan2f(cn, dt);
}

// ---------------------------------------------------------------------------
// (T,3) @ (3,64) + b  -- K=3 too small for WMMA
// ---------------------------------------------------------------------------
__global__ void k_geo_lin0(const float* __restrict__ geo3, const float* __restrict__ W,
                           const float* __restrict__ b, float* __restrict__ Y, int T) {
  long long idx = (long long)blockIdx.x * blockDim.x + threadIdx.x;
  if (idx >= (long long)T * HDIM) return;
  int c = (int)(idx & (HDIM - 1));
  long long r = idx >> 6;
  float g0 = geo3[3 * r], g1 = geo3[3 * r + 1], g2 = geo3[3 * r + 2];
  Y[idx] = b[c] + g0 * W[c] + g1 * W[HDIM + c] + g2 * W[2 * HDIM + c];
}

// ---------------------------------------------------------------------------
// Fused masked-BN-stats epilogue: wave reduces its 16x64 strip (in acc) and
// atomically accumulates per-channel sum/sumsq. branch<0 => all rows.
// ---------------------------------------------------------------------------
__device__ __forceinline__ void stats_epilogue(const v8f acc[4],
                                               const int* __restrict__ maskId,
                                               int branch, float* __restrict__ stat,
                                               int row0, int lh, int hi) {
  float mw[8];
#pragma unroll
  for (int v = 0; v < 8; ++v) {
    if (branch >= 0)
      mw[v] = (maskId[row0 + v + (hi << 3)] == branch) ? 1.f : 0.f;
    else
      mw[v] = 1.f;
  }
#pragma unroll
  for (int ct = 0; ct < 4; ++ct) {
    float s = 0.f, s2 = 0.f;
#pragma unroll
    for (int v = 0; v < 8; ++v) {
      float y = acc[ct][v];
      s += mw[v] * y;
      s2 += mw[v] * y * y;
    }
    s  += __shfl_xor(s, 16, 32);
    s2 += __shfl_xor(s2, 16, 32);
    if (hi == 0) {
      atomicAdd(&stat[ct * 16 + lh], s);
      atomicAdd(&stat[HDIM + ct * 16 + lh], s2);
    }
  }
}

// ---------------------------------------------------------------------------
// Fused WMMA GEMM: Y = relu(bn(Yprev)) @ W + bias, with masked stats of the
// *input* layer applied on the A-load path (scale/shift table in LDS) and
// masked stats of the *output* accumulated by the epilogue.
// One wave -> 16x64 strip. 8 waves / 256-thread block.
// ---------------------------------------------------------------------------
__global__ void k_gemm64_fused(const float* __restrict__ Yprev,
                               const float* __restrict__ pstat,
                               const float* __restrict__ pgamma,
                               const float* __restrict__ pbeta,
                               int pbranch,
                               const float* __restrict__ W,
                               const float* __restrict__ bias,
                               float* __restrict__ Yout,
                               const int* __restrict__ maskId, int branch,
                               float* __restrict__ stat,
                               const int* __restrict__ cnts, int T) {
  __shared__ float sW[32 * WSTRIDE];                 // 20 KB
  __shared__ __align__(16) float sSS[2 * HDIM];      // scale/shift pairs
  for (int idx = threadIdx.x; idx < HDIM * HDIM; idx += 256) {
    int r = idx >> 6, c = idx & 63;
    sW[(r >> 1) * WSTRIDE + c * 2 + (r & 1)] = W[idx];
  }
  if (threadIdx.x < HDIM) {
    int c = threadIdx.x;
    float cnt = (pbranch < 0) ? (float)T : fmaxf((float)cnts[pbranch], 1.f);
    float mean = pstat[c] / cnt;
    float var = pstat[HDIM + c] / cnt - mean * mean;
    float inv = rsqrtf(var + EPSV) * pgamma[c];
    sSS[2 * c]     = inv;
    sSS[2 * c + 1] = pbeta[c] - mean * inv;
  }
  __syncthreads();

  int rowTile = blockIdx.x * 8 + (threadIdx.x >> 5);
  if (rowTile >= (T >> 4)) return;
  int lane = threadIdx.x & 31;
  int lh = lane & 15;
  int hi = lane >> 4;
  int row0 = rowTile << 4;

  v8f acc[4];
#pragma unroll
  for (int ct = 0; ct < 4; ++ct) {
    float bv = bias[ct * 16 + lh];
#pragma unroll
    for (int v = 0; v < 8; ++v) acc[ct][v] = bv;
  }

  const float* xrow = Yprev + (long long)(row0 + lh) * HDIM;
#pragma unroll
  for (int k = 0; k < HDIM; k += 4) {
    int ka = k + (hi << 1);
    v2f yr;
    yr.x = xrow[ka];
    yr.y = xrow[ka + 1];
    v4f ss = *(const v4f*)&sSS[ka * 2];  // broadcast within half-wave
    v2f a;
    a.x = fmaxf(fmaf(yr.x, ss.x, ss.y), 0.f);
    a.y = fmaxf(fmaf(yr.y, ss.z, ss.w), 0.f);
    int k2 = (k >> 1) + hi;
#pragma unroll
    for (int ct = 0; ct < 4; ++ct) {
      v2f b = *(const v2f*)&sW[k2 * WSTRIDE + (ct * 16 + lh) * 2];
      acc[ct] = wmma4(a, b, acc[ct]);
    }
  }
#pragma unroll
  for (int ct = 0; ct < 4; ++ct)
#pragma unroll
    for (int v = 0; v < 8; ++v)
      Yout[(long long)(row0 + v + (hi << 3)) * HDIM + ct * 16 + lh] = acc[ct][v];

  stats_epilogue(acc, maskId, branch, stat, row0, lh, hi);
}

// ---------------------------------------------------------------------------
// WMMA GEMM layer 0 + fused output stats:
// Y(T,64) = [nf[i] | nf[j] | nf[k] | geo](T,256) @ W(256,64) + b
// Rows gathered on the fly (no 102MB xcat). Swizzled W = 80 KB LDS (<320KB/WGP).
// ---------------------------------------------------------------------------
__global__ void k_gemm_l0(const float* __restrict__ NF, const float* __restrict__ GEO,
                          const int* __restrict__ ti, const int* __restrict__ tj,
                          const int* __restrict__ tk,
                          const float* __restrict__ W, const float* __restrict__ bias,
                          float* __restrict__ Y,
                          const int* __restrict__ maskId, int branch,
                          float* __restrict__ stat, int T) {
  __shared__ float sW[128 * WSTRIDE];  // 80 KB
  for (int idx = threadIdx.x; idx < 4 * HDIM * HDIM; idx += 256) {
    int r = idx >> 6, c = idx & 63;
    sW[(r >> 1) * WSTRIDE + c * 2 + (r & 1)] = W[idx];
  }
  __syncthreads();

  int rowTile = blockIdx.x * 8 + (threadIdx.x >> 5);
  if (rowTile >= (T >> 4)) return;
  int lane = threadIdx.x & 31;
  int lh = lane & 15;
  int hi = lane >> 4;
  int row0 = rowTile << 4;
  int arow = row0 + lh;

  const float* seg0 = NF + (long long)ti[arow] * HDIM;
  const float* seg1 = NF + (long long)tj[arow] * HDIM;
  const float* seg2 = NF + (long long)tk[arow] * HDIM;
  const float* seg3 = GEO + (long long)arow * HDIM;

  v8f acc[4];
#pragma unroll
  for (int ct = 0; ct < 4; ++ct) {
    float bv = bias[ct * 16 + lh];
#pragma unroll
    for (int v = 0; v < 8; ++v) acc[ct][v] = bv;
  }

#pragma unroll 4
  for (int k = 0; k < 4 * HDIM; k += 4) {
    const float* src = (k < HDIM) ? seg0
                     : (k < 2 * HDIM) ? seg1
                     : (k < 3 * HDIM) ? seg2 : seg3;
    int ka = (k & (HDIM - 1)) + (hi << 1);
    v2f a;
    a.x = src[ka];
    a.y = src[ka + 1];
    int k2 = (k >> 1) + hi;
#pragma unroll
    for (int ct = 0; ct < 4; ++ct) {
      v2f b = *(const v2f*)&sW[k2 * WSTRIDE + (ct * 16 + lh) * 2];
      acc[ct] = wmma4(a, b, acc[ct]);
    }
  }
#pragma unroll
  for (int ct = 0; ct < 4; ++ct)
#pragma unroll
    for (int v = 0; v < 8; ++v)
      Y[(long long)(row0 + v + (hi << 3)) * HDIM + ct * 16 + lh] = acc[ct][v];

  stats_epilogue(acc, maskId, branch, stat, row0, lh, hi);
}

// ---------------------------------------------------------------------------
// Standalone masked BN stats (only for the non-WMMA geo_lin0 layer).
// ---------------------------------------------------------------------------
__global__ void k_bn_stats(const float* __restrict__ Y, const int* __restrict__ maskId,
                           int branch, float* __restrict__ stat, int T) {
  int c = threadIdx.x & (HDIM - 1);
  int rg = threadIdx.x >> 6;
  float s = 0.f, s2 = 0.f;
  for (int r = blockIdx.x * 4 + rg; r < T; r += gridDim.x * 4) {
    bool m = (branch < 0) || (maskId[r] == branch);
    float v = Y[(long long)r * HDIM + c];
    if (m) { s += v; s2 += v * v; }
  }
  __shared__ float sh0[256];
  __shared__ float sh1[256];
  sh0[threadIdx.x] = s;
  sh1[threadIdx.x] = s2;
  __syncthreads();
  if (threadIdx.x < HDIM) {
    float t1 = sh0[c] + sh0[c + 64] + sh0[c + 128] + sh0[c + 192];
    float t2 = sh1[c] + sh1[c + 64] + sh1[c + 128] + sh1[c + 192];
    atomicAdd(&stat[c], t1);
    atomicAdd(&stat[HDIM + c], t2);
  }
}

// ---------------------------------------------------------------------------
// BN normalize + relu (materializes activated geo features geoH).
// ---------------------------------------------------------------------------
__global__ void k_bn_apply(const float* __restrict__ Y, const float* __restrict__ stat,
                           const float* __restrict__ gamma, const float* __restrict__ beta,
                           float* __restrict__ Xo, int T) {
  long long idx = (long long)blockIdx.x * blockDim.x + threadIdx.x;
  if (idx >= (long long)T * HDIM) return;
  int c = (int)(idx & (HDIM - 1));
  float cnt = (float)T;
  float mean = stat[c] / cnt;
  float var = stat[HDIM + c] / cnt - mean * mean;
  float inv = rsqrtf(var + EPSV) * gamma[c];
  float z = (Y[idx] - mean) * inv + beta[c];
  Xo[idx] = fmaxf(z, 0.f);
}

// ---------------------------------------------------------------------------
// Final layer fused: bn + relu (leaky==identity) * att, masked rows only,
// scattered directly into out via segment-sum atomics (masks partition rows,
// so branch contributions never overlap on a row).
// ---------------------------------------------------------------------------
__global__ void k_final_seg(const float* __restrict__ Y, const float* __restrict__ stat,
                            const float* __restrict__ gamma, const float* __restrict__ beta,
                            const int* __restrict__ cnts, const int* __restrict__ maskId,
                            const float* __restrict__ attp, int branch,
                            const int* __restrict__ ti, float* __restrict__ out, int T) {
  long long idx = (long long)blockIdx.x * blockDim.x + threadIdx.x;
  if (idx >= (long long)T * HDIM) return;
  long long r = idx >> 6;
  if (maskId[r] != branch) return;
  int c = (int)(idx & (HDIM - 1));
  float cnt = fmaxf((float)cnts[branch], 1.f);
  float mean = stat[c] / cnt;
  float var = stat[HDIM + c] / cnt - mean * mean;
  float inv = rsqrtf(var + EPSV) * gamma[c];
  float z = fmaxf((Y[idx] - mean) * inv + beta[c], 0.f) * attp[branch];
  atomicAdd(&out[(long long)ti[r] * HDIM + c], z);
}

// ---------------------------------------------------------------------------
// mask ids + per-branch counts; num_edge_inside read on-device
// ---------------------------------------------------------------------------
__global__ void k_masks(const int* __restrict__ edx_ij, const int* __restrict__ edx_jk,
                        const int* __restrict__ nei_p, int* __restrict__ maskId,
                        int* __restrict__ cnts, int T) {
  int r = blockIdx.x * blockDim.x + threadIdx.x;
  if (r >= T) return;
  int nei = *nei_p;
  int id = ((edx_ij[r] < nei) ? 0 : 2) + ((edx_jk[r] < nei) ? 0 : 1);
  maskId[r] = id;
  atomicAdd(&cnts[id], 1);
}

// ===========================================================================
extern "C" void kernel_launch(void* const* d_in, const int* in_sizes, int n_in,
                              void* d_out, int out_size, void* d_ws, size_t ws_size,
                              hipStream_t stream) {
  const float* input_feature = (const float*)d_in[0];
  const float* coords        = (const float*)d_in[1];
  const int*   e2rd          = (const int*)d_in[3];
  const int*   edx_jk        = (const int*)d_in[4];
  const int*   edx_ij        = (const int*)d_in[5];
  const int*   nei_p         = (const int*)d_in[7];

  const int T = in_sizes[4];            // 100000 (multiple of 16)
  const int N = in_sizes[0] / HDIM;     // 50000
  const int* ti = e2rd;
  const int* tj = e2rd + T;
  const int* tk = e2rd + 2 * T;

  // params: pytree leaves, dict keys sorted: att, mlp_geo, wmlp; layers W,b,beta,gamma
  struct LayerP { const float *W, *b, *beta, *gamma; };
  int p = 9;
  const float* att = (const float*)d_in[p++];
  LayerP geoL[2];
  for (int l = 0; l < 2; ++l) {
    geoL[l].W     = (const float*)d_in[p++];
    geoL[l].b     = (const float*)d_in[p++];
    geoL[l].beta  = (const float*)d_in[p++];
    geoL[l].gamma = (const float*)d_in[p++];
  }
  LayerP wm[3][4][4];
  for (int t = 0; t < 3; ++t)
    for (int br = 0; br < 4; ++br)
      for (int l = 0; l < 4; ++l) {
        wm[t][br][l].W     = (const float*)d_in[p++];
        wm[t][br][l].b     = (const float*)d_in[p++];
        wm[t][br][l].beta  = (const float*)d_in[p++];
        wm[t][br][l].gamma = (const float*)d_in[p++];
      }

  // workspace layout (xout eliminated)
  const int NSLOT = 50;
  float* ws    = (float*)d_ws;
  float* geo3  = ws;
  float* bufA  = geo3 + (size_t)3 * T;
  float* bufB  = bufA + (size_t)T * HDIM;
  float* geoH  = bufB + (size_t)T * HDIM;
  float* stats = geoH + (size_t)T * HDIM;
  int*   cnts  = (int*)(stats + (size_t)NSLOT * 2 * HDIM);
  int*   maskId = cnts + 4;

  hipMemsetAsync(stats, 0, (size_t)NSLOT * 2 * HDIM * sizeof(float) + 4 * sizeof(int),
                 stream);
  hipMemsetAsync(d_out, 0, (size_t)out_size * sizeof(float), stream);

  const int BT = 256;
  const int gT    = (T + BT - 1) / BT;
  const int gTH   = (int)(((long long)T * HDIM + BT - 1) / BT);
  const int rowTiles = T / 16;
  const int gGemm = (rowTiles + 7) / 8;  // 1 wave per 16x64 strip, 8 waves/block
  const int gStat = 512;

  // masks first (geo layers pass branch=-1 and never read maskId)
  k_masks<<<gT, BT, 0, stream>>>(edx_ij, edx_jk, nei_p, maskId, cnts, T);

  // geometry MLP: lin0 -> stats -> fused gemm (normalizes lin0 preact on load)
  k_geo_feats<<<gT, BT, 0, stream>>>(coords, ti, tj, tk, geo3, T);
  k_geo_lin0<<<gTH, BT, 0, stream>>>(geo3, geoL[0].W, geoL[0].b, bufA, T);
  int slot = 0;
  k_bn_stats<<<gStat, BT, 0, stream>>>(bufA, maskId, -1, stats + slot * 2 * HDIM, T);
  float* s0 = stats + slot * 2 * HDIM; slot++;
  float* s1 = stats + slot * 2 * HDIM; slot++;
  k_gemm64_fused<<<gGemm, BT, 0, stream>>>(bufA, s0, geoL[0].gamma, geoL[0].beta, -1,
                                           geoL[1].W, geoL[1].b, bufB,
                                           maskId, -1, s1, cnts, T);
  k_bn_apply<<<gTH, BT, 0, stream>>>(bufB, s1, geoL[1].gamma, geoL[1].beta, geoH, T);

  // interactions: per branch, gemm_l0 -> 3x fused gemm -> fused final+segsum
  const float* nodeF = input_feature;
  for (int t = 0; t < 3; ++t) {
    float* outSlice = (float*)d_out + (size_t)t * N * HDIM;
    for (int br = 0; br < 4; ++br) {
      float* sl[4];
      for (int l = 0; l < 4; ++l) { sl[l] = stats + slot * 2 * HDIM; slot++; }

      k_gemm_l0<<<gGemm, BT, 0, stream>>>(nodeF, geoH, ti, tj, tk,
                                          wm[t][br][0].W, wm[t][br][0].b, bufA,
                                          maskId, br, sl[0], T);
      k_gemm64_fused<<<gGemm, BT, 0, stream>>>(bufA, sl[0], wm[t][br][0].gamma,
                                               wm[t][br][0].beta, br,
                                               wm[t][br][1].W, wm[t][br][1].b, bufB,
                                               maskId, br, sl[1], cnts, T);
      k_gemm64_fused<<<gGemm, BT, 0, stream>>>(bufB, sl[1], wm[t][br][1].gamma,
                                               wm[t][br][1].beta, br,
                                               wm[t][br][2].W, wm[t][br][2].b, bufA,
                                               maskId, br, sl[2], cnts, T);
      k_gemm64_fused<<<gGemm, BT, 0, stream>>>(bufA, sl[2], wm[t][br][2].gamma,
                                               wm[t][br][2].beta, br,
                                               wm[t][br][3].W, wm[t][br][3].b, bufB,
                                               maskId, br, sl[3], cnts, T);
      k_final_seg<<<gTH, BT, 0, stream>>>(bufB, sl[3], wm[t][br][3].gamma,
                                          wm[t][br][3].beta, cnts, maskId, att, br,
                                          ti, outSlice, T);
    }
    nodeF = outSlice;  // node_feature for next interaction
  }
}